// Selector_75917841924359
// MI455X (gfx1250) — compile-verified
//
#include <hip/hip_runtime.h>
#include <math.h>

#define C_DIM 512
#define H_DIM 192
#define W_DIM 336
#define HW    (H_DIM * W_DIM)   // 64512
#define P_BOX 2000
#define SORT_N 2048

typedef __attribute__((ext_vector_type(2))) float v2f;
typedef __attribute__((ext_vector_type(8))) float v8f;

// ---------------------------------------------------------------------------
// Kernel 1: channel-mean map g[p] = (1/C) * sum_c x[c*HW + p]
// One wave produces 16 pixels using V_WMMA_F32_16X16X4_F32 with A = ones(16x4):
//   D[m][n] = sum_k B[k][n] + C[m][n]  -> per-column (per-pixel) channel sums.
// B (4x16, K x N) VGPR layout: VGPR0 = rows K=0 (lanes 0-15) / K=2 (lanes 16-31),
//                              VGPR1 = rows K=1 / K=3.
// So lane l loads channels (c+kbase, c+kbase+1) at pixel pix0 + (l&15), where
// kbase = 0 for lanes 0-15 and 2 for lanes 16-31. Each load instruction covers
// two contiguous 64B segments -> coalesced streaming of the full 132 MB tensor.
// D row 0 (acc VGPR0, lanes 0-15) holds the 16 pixel sums.
// ---------------------------------------------------------------------------
__global__ void __launch_bounds__(256) chanmean_wmma(const float* __restrict__ x,
                                                     float* __restrict__ g) {
  const int lane        = threadIdx.x & 31;
  const int waveInBlock = threadIdx.x >> 5;
  const int gwave       = blockIdx.x * 8 + waveInBlock;
  const int pix0        = gwave * 16;
  const int colN        = lane & 15;
  const int kbase       = (lane < 16) ? 0 : 2;

  const float* p0 = x + (size_t)kbase * HW + pix0 + colN;  // channel c+kbase
  const float* p1 = p0 + HW;                               // channel c+kbase+1

  v2f a; a.x = 1.0f; a.y = 1.0f;   // A = ones(16x4), layout-independent
  v8f acc = {};

#pragma unroll 4
  for (int c = 0; c < C_DIM; c += 4) {
    v2f b;
    b.x = p0[0];
    b.y = p1[0];
    acc = __builtin_amdgcn_wmma_f32_16x16x4_f32(
        /*neg_a=*/false, a, /*neg_b=*/false, b,
        /*c_mod=*/(short)0, acc, /*reuse_a=*/false, /*reuse_b=*/false);
    p0 += 4 * HW;
    p1 += 4 * HW;
  }

  if (lane < 16) g[pix0 + colN] = acc[0] * (1.0f / (float)C_DIM);
}

// ---------------------------------------------------------------------------
// Kernel 2: in-place 2D inclusive prefix sum (summed-area table) over g[H,W].
// Single workgroup; row scans then column scans with a device fence between.
// ---------------------------------------------------------------------------
__global__ void __launch_bounds__(384) sat_inplace(float* __restrict__ g) {
  const int t = threadIdx.x;
  if (t < H_DIM) {
    float s = 0.0f;
    float* row = g + t * W_DIM;
    for (int j = 0; j < W_DIM; ++j) { s += row[j]; row[j] = s; }
  }
  __threadfence();
  __syncthreads();
  if (t < W_DIM) {
    float s = 0.0f;
    for (int i = 0; i < H_DIM; ++i) {
      s += g[i * W_DIM + t];
      g[i * W_DIM + t] = s;
    }
  }
}

// ---------------------------------------------------------------------------
// Kernel 3: per-box mean score from the SAT (inclusive; index -1 == 0).
// ---------------------------------------------------------------------------
__device__ __forceinline__ float sat_at(const float* sat, int y, int x) {
  return (y >= 0 && x >= 0) ? sat[y * W_DIM + x] : 0.0f;
}

__global__ void box_scores(const float* __restrict__ sat,
                           const float* __restrict__ bbox,
                           const float* __restrict__ scale,
                           float* __restrict__ scores) {
  const int p = blockIdx.x * blockDim.x + threadIdx.x;
  if (p >= P_BOX) return;
  const float sx = scale[1];
  const float sy = scale[0];
  int x1 = min(max((int)floorf(bbox[p * 4 + 0] / sx), 0), W_DIM - 1);
  int y1 = min(max((int)floorf(bbox[p * 4 + 1] / sy), 0), H_DIM - 1);
  int x2 = min(max((int)floorf(bbox[p * 4 + 2] / sx), 0), W_DIM - 1);
  int y2 = min(max((int)floorf(bbox[p * 4 + 3] / sy), 0), H_DIM - 1);
  float bs = sat_at(sat, y2, x2) - sat_at(sat, y1 - 1, x2)
           - sat_at(sat, y2, x1 - 1) + sat_at(sat, y1 - 1, x1 - 1);
  float area = (float)((y2 - y1 + 1) * (x2 - x1 + 1));
  scores[p] = bs / area;
}

// ---------------------------------------------------------------------------
// Kernel 4: deterministic bitonic sort (ascending) of 2048 padded scores in
// LDS, derive threshold between sn-th and (sn+1)-th largest, write sigmoid.
// ---------------------------------------------------------------------------
__global__ void __launch_bounds__(1024) topk_sigmoid(const float* __restrict__ scores,
                                                     const int* __restrict__ seln,
                                                     float* __restrict__ out) {
  __shared__ float sv[SORT_N];
  const int t = threadIdx.x;
  for (int i = t; i < SORT_N; i += 1024)
    sv[i] = (i < P_BOX) ? scores[i] : -INFINITY;
  __syncthreads();

  for (int k = 2; k <= SORT_N; k <<= 1) {
    for (int j = k >> 1; j > 0; j >>= 1) {
      for (int i = t; i < SORT_N; i += 1024) {
        int ixj = i ^ j;
        if (ixj > i) {
          bool asc = ((i & k) == 0);
          float a = sv[i];
          float b = sv[ixj];
          if ((a > b) == asc) { sv[i] = b; sv[ixj] = a; }
        }
      }
      __syncthreads();
    }
  }

  const int sn = *seln;  // select_num (10)
  // descending top[i] == ascending sv[SORT_N-1-i]
  const float thresh = 0.5f * (sv[SORT_N - sn] + sv[SORT_N - 1 - sn]);

  for (int i = t; i < P_BOX; i += 1024) {
    float z = (scores[i] - thresh) * 100.0f;
    out[i] = 1.0f / (1.0f + expf(-z));
  }
}

// ---------------------------------------------------------------------------
extern "C" void kernel_launch(void* const* d_in, const int* in_sizes, int n_in,
                              void* d_out, int out_size, void* d_ws, size_t ws_size,
                              hipStream_t stream) {
  const float* x     = (const float*)d_in[0];   // [1,512,192,336] f32
  const float* bbox  = (const float*)d_in[1];   // [2000,4] f32
  const float* scale = (const float*)d_in[2];   // [2] f32
  const int*   seln  = (const int*)d_in[3];     // scalar int (10)
  float*       out   = (float*)d_out;           // 2000 f32

  float* g      = (float*)d_ws;   // HW floats; becomes the SAT in place
  float* scores = g + HW;         // P_BOX floats

  // 1) channel mean via WMMA: 504 blocks x 8 waves x 16 pixels = 64512 pixels
  chanmean_wmma<<<HW / (16 * 8), 256, 0, stream>>>(x, g);
  // 2) summed-area table in place
  sat_inplace<<<1, 384, 0, stream>>>(g);
  // 3) per-box scores
  box_scores<<<(P_BOX + 255) / 256, 256, 0, stream>>>(g, bbox, scale, scores);
  // 4) threshold + sigmoid
  topk_sigmoid<<<1, 1024, 0, stream>>>(scores, seln, out);
}